// latentAttentionGain_45397804319311
// MI455X (gfx1250) — compile-verified
//
#include <hip/hip_runtime.h>
#include <math.h>

typedef float v2f __attribute__((ext_vector_type(2)));
typedef float v8f __attribute__((ext_vector_type(8)));

#define BS 32
#define EN 256
#define DX 512
#define DZ 256
#define L_TOT 1280
#define NK 1024   // keys computed: global key kk = 256 + k
#define NH 8

// workspace layout (float offsets)
#define WS_KP     0                          // [BS][NK][EN]
#define WS_MEAN   (WS_KP + BS*NK*EN)         // [BS][NK]
#define WS_QW2    (WS_MEAN + BS*NK)          // [256][EN]  rows 256..511 of Q@Wq
#define WS_QCONST (WS_QW2 + 256*EN)          // [BS][EN]   pe@Wq + bq
#define WS_KCONST (WS_QCONST + BS*EN)        // [BS][EN]   pe@Wk + bk
#define WS_SUMWK  (WS_KCONST + BS*EN)        // [EN]       colsum(Wk)
#define WS_W4     (WS_SUMWK + EN)            // [BS][256][4]

// ---------------- zero the atten output region ----------------
__global__ void zero_atten_kernel(float4* atten4, long n4) {
  long i = (long)blockIdx.x * blockDim.x + threadIdx.x;
  long stride = (long)gridDim.x * blockDim.x;
  float4 z; z.x = 0.f; z.y = 0.f; z.z = 0.f; z.w = 0.f;
  for (long k = i; k < n4; k += stride) atten4[k] = z;
}

// atten[b,q,q] = 1 for q < 256 (one-hot rows)
__global__ void diag_ones_kernel(float* atten) {
  int i = blockIdx.x * blockDim.x + threadIdx.x;   // 0..8191
  int b = i >> 8, q = i & 255;
  atten[(size_t)b * DX * L_TOT + (size_t)q * L_TOT + q] = 1.0f;
}

// ---------------- per-column means over ensemble axis ----------------
// block = (b, 64-k chunk); 256 threads = 64 k x 4 e-phases, coalesced along k
__global__ void means_kernel(const float* state, const float* obs0,
                             const float* obs1, const float* obs2, float* ws) {
  __shared__ float red[4][64];
  int blk = blockIdx.x;           // 0..511
  int b  = blk >> 4;
  int k0 = (blk & 15) * 64;
  int t = threadIdx.x;
  int kl = t & 63;
  int eo = t >> 6;                // 0..3
  int kk = 256 + k0 + kl;
  const float* base; int rs;
  if (kk < 512) { base = state + (size_t)b * EN * DX + kk; rs = DX; }
  else {
    int m = (kk - 512) >> 8;
    const float* o = (m == 0) ? obs0 : ((m == 1) ? obs1 : obs2);
    base = o + (size_t)b * EN * DZ + ((kk - 512) & 255); rs = DZ;
  }
  float s = 0.f;
  for (int e = eo; e < EN; e += 4) s += base[(size_t)e * rs];
  red[eo][kl] = s;
  __syncthreads();
  if (t < 64) {
    float m = (red[0][t] + red[1][t] + red[2][t] + red[3][t]) * (1.0f / EN);
    ws[WS_MEAN + b * NK + k0 + t] = m;
  }
}

// ---------------- QW2[i][j] = sum_e Q[256+i,e]*Wq[e,j] ----------------
__global__ void qw2_kernel(const float* Q, const float* Wq, float* ws) {
  __shared__ float qrow[EN];
  int i = blockIdx.x, j = threadIdx.x;
  qrow[j] = Q[(size_t)(256 + i) * EN + j];
  __syncthreads();
  float acc = 0.f;
  for (int e = 0; e < EN; ++e) acc = fmaf(qrow[e], Wq[(size_t)e * EN + j], acc);
  ws[WS_QW2 + i * EN + j] = acc;
}

// ---------------- pe@Wq + bq and pe@Wk + bk (pe built on-device) ----------------
__global__ void peproj_kernel(const float* Wq, const float* bq,
                              const float* Wk, const float* bk, float* ws) {
  __shared__ float pe[EN];
  int b = blockIdx.x, j = threadIdx.x;
  const float C = 9.210340371976184f / 256.0f;   // ln(10000)/d_model
  int ebase = j & ~1;
  float f = expf(-(float)ebase * C);
  float ang = (float)b * f;
  pe[j] = (j & 1) ? cosf(ang) : sinf(ang);
  __syncthreads();
  float aq = 0.f, ak = 0.f;
  for (int e = 0; e < EN; ++e) {
    float p = pe[e];
    aq = fmaf(p, Wq[(size_t)e * EN + j], aq);
    ak = fmaf(p, Wk[(size_t)e * EN + j], ak);
  }
  ws[WS_QCONST + b * EN + j] = aq + bq[j];
  ws[WS_KCONST + b * EN + j] = ak + bk[j];
}

// ---------------- sumWk[j] = sum_e Wk[e,j] ----------------
__global__ void sumwk_kernel(const float* Wk, float* ws) {
  int j = threadIdx.x;
  float s = 0.f;
  for (int e = 0; e < EN; ++e) s += Wk[(size_t)e * EN + j];
  ws[WS_SUMWK + j] = s;
}

// ---------------- K projection GEMM via V_WMMA_F32_16X16X4_F32 ----------------
// Kp[b,k,j] = sum_e In[b,k,e]*Wk[e,j] + kconst[b,j] - mean[b,k]*sumWk[j]
// In[b,k,e] = (state|obs_m) column (kk = 256+k), transposed read (k-contiguous -> coalesced)
// grid (jt=4, kt=16, b=32), 128 threads = 4 waves; WG computes 64x64, wave = 16x64
__global__ void kproj_wmma_kernel(const float* state, const float* obs0,
                                  const float* obs1, const float* obs2,
                                  const float* Wk, float* ws) {
  __shared__ float As[16][65];   // [e][k]
  __shared__ float Bs[16][66];   // [e][j]
  int jt = blockIdx.x, kt = blockIdx.y, b = blockIdx.z;
  int k0 = kt * 64, j0 = jt * 64;
  int t = threadIdx.x;
  int w = t >> 5, lane = t & 31;
  int lm = lane & 15;
  int hi = lane >> 4;            // 0: lanes 0-15, 1: lanes 16-31

  int kk0 = 256 + k0;
  const float* abase; int ars;
  if (kk0 < 512) { abase = state + (size_t)b * EN * DX + kk0; ars = DX; }
  else {
    int m = (kk0 - 512) >> 8;
    const float* o = (m == 0) ? obs0 : ((m == 1) ? obs1 : obs2);
    abase = o + (size_t)b * EN * DZ + ((kk0 - 512) & 255); ars = DZ;
  }

  v8f acc[4] = {};
  for (int e0 = 0; e0 < EN; e0 += 16) {
    #pragma unroll
    for (int i = 0; i < 8; ++i) {              // 64k x 16e tile, k-contiguous loads
      int idx = i * 128 + t;
      int kl = idx & 63, el = idx >> 6;
      As[el][kl] = abase[(size_t)(e0 + el) * ars + kl];
    }
    #pragma unroll
    for (int i = 0; i < 8; ++i) {              // 16e x 64j tile, j-contiguous loads
      int idx = i * 128 + t;
      int jl = idx & 63, el = idx >> 6;
      Bs[el][jl] = Wk[(size_t)(e0 + el) * EN + j0 + jl];
    }
    __syncthreads();
    #pragma unroll
    for (int s = 0; s < 4; ++s) {              // K substeps of 4
      v2f a;
      a.x = As[s * 4 + hi * 2 + 0][w * 16 + lm];   // A: M=lane&15, K per VGPR/half
      a.y = As[s * 4 + hi * 2 + 1][w * 16 + lm];
      #pragma unroll
      for (int js = 0; js < 4; ++js) {
        v2f bf;
        bf.x = Bs[s * 4 + hi * 2 + 0][js * 16 + lm]; // B: N=lane&15, K per VGPR/half
        bf.y = Bs[s * 4 + hi * 2 + 1][js * 16 + lm];
        acc[js] = __builtin_amdgcn_wmma_f32_16x16x4_f32(
            false, a, false, bf, (short)0, acc[js], false, false);
      }
    }
    __syncthreads();
  }

  // fused epilogue: + (pe@Wk + bk) - mean * colsum(Wk), then store Kp
  #pragma unroll
  for (int js = 0; js < 4; ++js) {
    int col = j0 + js * 16 + lm;
    float kc = ws[WS_KCONST + b * EN + col];
    float sw = ws[WS_SUMWK + col];
    #pragma unroll
    for (int v = 0; v < 8; ++v) {
      int row = k0 + w * 16 + hi * 8 + v;      // D: M = vgpr + 8*half
      float mean = ws[WS_MEAN + b * NK + row];
      ws[WS_KP + ((size_t)b * NK + row) * EN + col] = acc[js][v] + kc - mean * sw;
    }
  }
}

// ---------------- scores -> per-head softmax(4) -> head-avg weights ----------------
// block = (b, qi) with q = 256+qi; 256 threads, wave h == head h (dh == 32 lanes)
__global__ void weights_kernel(float* ws, float* atten) {
  __shared__ float hw[NH][4];
  int blk = blockIdx.x;
  int b = blk >> 8, qi = blk & 255;
  int j = threadIdx.x;
  int h = j >> 5, lane = j & 31;
  float qv = ws[WS_QW2 + qi * EN + j] + ws[WS_QCONST + b * EN + j];
  const float* kp = ws + WS_KP + ((size_t)b * NK + qi) * EN;  // key rows t*256+qi
  float s[4];
  #pragma unroll
  for (int tt = 0; tt < 4; ++tt) {
    float p = qv * kp[(size_t)tt * 256 * EN + j];
    for (int m = 16; m >= 1; m >>= 1) p += __shfl_xor(p, m, 32);
    s[tt] = p;
  }
  if (lane == 0) {
    const float sc = 0.17677669529663687f;     // 1/sqrt(dh=32)
    float st[4]; float mx = -1e30f;
    #pragma unroll
    for (int tt = 0; tt < 4; ++tt) { st[tt] = s[tt] * sc; mx = fmaxf(mx, st[tt]); }
    float sum = 0.f;
    #pragma unroll
    for (int tt = 0; tt < 4; ++tt) { st[tt] = expf(st[tt] - mx); sum += st[tt]; }
    float inv = 1.0f / sum;
    #pragma unroll
    for (int tt = 0; tt < 4; ++tt) hw[h][tt] = st[tt] * inv;
  }
  __syncthreads();
  if (j < 4) {
    float wsum = 0.f;
    #pragma unroll
    for (int h2 = 0; h2 < NH; ++h2) wsum += hw[h2][j];
    wsum *= 0.125f;                            // mean over heads
    ws[WS_W4 + ((size_t)b * 256 + qi) * 4 + j] = wsum;
    int q = 256 + qi;                          // atten cols: q + 256*t
    atten[(size_t)b * DX * L_TOT + (size_t)q * L_TOT + q + 256 * j] = wsum;
  }
}

// ---------------- new_state, fully coalesced along q ----------------
// block = (b, e); thread t handles q=t (copy) and q=256+t (4-term weighted sum)
__global__ void newstate_kernel(const float* state, const float* obs0,
                                const float* obs1, const float* obs2,
                                const float* ws, float* out_ns) {
  int blk = blockIdx.x;
  int b = blk >> 8, e = blk & 255;
  int t = threadIdx.x;
  const float* srow = state + ((size_t)b * EN + e) * DX;
  float* orow = out_ns + ((size_t)b * EN + e) * DX;
  orow[t] = srow[t];                           // q < 256: one-hot -> copy
  const float4 w4 = *(const float4*)(ws + WS_W4 + ((size_t)b * 256 + t) * 4);
  size_t oo = ((size_t)b * EN + e) * DZ + t;
  float v = w4.x * srow[256 + t] + w4.y * obs0[oo] + w4.z * obs1[oo] + w4.w * obs2[oo];
  orow[256 + t] = v;
}

extern "C" void kernel_launch(void* const* d_in, const int* in_sizes, int n_in,
                              void* d_out, int out_size, void* d_ws, size_t ws_size,
                              hipStream_t stream) {
  const float* state = (const float*)d_in[0];
  const float* obs0  = (const float*)d_in[1];
  const float* obs1  = (const float*)d_in[2];
  const float* obs2  = (const float*)d_in[3];
  const float* Q     = (const float*)d_in[4];
  const float* Wq    = (const float*)d_in[5];
  const float* bq    = (const float*)d_in[6];
  const float* Wk    = (const float*)d_in[7];
  const float* bk    = (const float*)d_in[8];
  (void)in_sizes; (void)n_in; (void)out_size; (void)ws_size;

  float* out_ns = (float*)d_out;                          // [32,256,512]
  float* atten  = out_ns + (size_t)BS * EN * DX;          // [32,512,1280]
  float* ws     = (float*)d_ws;

  long n4 = (long)BS * DX * L_TOT / 4;
  zero_atten_kernel<<<4096, 256, 0, stream>>>((float4*)atten, n4);
  diag_ones_kernel<<<32, 256, 0, stream>>>(atten);
  means_kernel<<<512, 256, 0, stream>>>(state, obs0, obs1, obs2, ws);
  qw2_kernel<<<256, 256, 0, stream>>>(Q, Wq, ws);
  peproj_kernel<<<32, 256, 0, stream>>>(Wq, bq, Wk, bk, ws);
  sumwk_kernel<<<1, 256, 0, stream>>>(Wk, ws);
  kproj_wmma_kernel<<<dim3(4, 16, 32), 128, 0, stream>>>(state, obs0, obs1, obs2, Wk, ws);
  weights_kernel<<<8192, 256, 0, stream>>>(ws, atten);
  newstate_kernel<<<8192, 256, 0, stream>>>(state, obs0, obs1, obs2, ws, out_ns);
}